// GlobalAggregator_57878979281136
// MI455X (gfx1250) — compile-verified
//
#include <hip/hip_runtime.h>
#include <cstddef>

typedef __attribute__((ext_vector_type(2))) float v2f;
typedef __attribute__((ext_vector_type(8))) float v8f;

#define DIM      128
#define S_NB     12
#define IT_WAVE  4          // items per wave
#define IT_WG    16         // items per workgroup (4 waves)
#define AP_STRIDE 34        // padded pair-row stride for A tiles in LDS
#define W1T_STRIDE 256      // pair-row stride for transposed w1 in global ws

// ---------------------------------------------------------------------------
// Pre-transpose w1 [129][128] into pair layout: w1t[k2*256 + col*2 + (k&1)]
// rows k2=0..65 ; k>=129 slots zero-filled so the WMMA K-loop is branchless.
// ---------------------------------------------------------------------------
__global__ void prep_w1_kernel(const float* __restrict__ w1,
                               float* __restrict__ w1t) {
    int idx = blockIdx.x * 256 + threadIdx.x;      // 66*256 = 16896 total
    if (idx >= 66 * 256) return;
    int k2  = idx >> 8;
    int rem = idx & 255;
    int col = rem >> 1;
    int k   = k2 * 2 + (rem & 1);
    w1t[idx] = (k < DIM + 1) ? w1[k * DIM + col] : 0.0f;
}

// ---------------------------------------------------------------------------
// aggregate: out[n] = sum_s softmax_s( lrelu((item[n]*nb[n,s] ++ nw[n,s]) @ w1) @ w2 ) * nb[n,s]
// grid = N_items/16 blocks, 128 threads (4 waves), wave handles 4 items.
// ---------------------------------------------------------------------------
__global__ __launch_bounds__(128)
void agg_kernel(const float* __restrict__ item_h,   // [N, 128]
                const float* __restrict__ nb_h,     // [N*12, 128]
                const float* __restrict__ nb_w,     // [N*12]
                const float* __restrict__ w1t,      // [66*256] pair layout
                const float* __restrict__ w2,       // [128]
                float* __restrict__ outp)           // [N, 128]
{
    __shared__ float ap[4][66 * AP_STRIDE];   // per-wave A-tile (pair layout)
    __shared__ float evals[4][48];            // attention logits
    __shared__ float alph[4][48];             // softmax weights

    const int lane = threadIdx.x & 31;
    const int wave = threadIdx.x >> 5;
    const int half = lane >> 4;               // K-half within 4-wide step
    const int l    = lane & 15;               // M (A) / N (B,C) index
    float* apw = &ap[wave][0];

    // zero the k2=65 pad row once (read by the bias step's half==1 lanes)
    for (int j = lane; j < AP_STRIDE; j += 32) apw[65 * AP_STRIDE + j] = 0.0f;

    // per-lane w2 values for each of the 8 N-tiles
    float w2r[8];
#pragma unroll
    for (int t = 0; t < 8; ++t) w2r[t] = w2[t * 16 + l];

    const int item0 = blockIdx.x * IT_WG + wave * IT_WAVE;

    for (int t = 0; t < 3; ++t) {             // 3 tiles of 16 rows = 48 rows
        const int rowbase = item0 * S_NB + t * 16;

        // ---- stage A tile: A[r][k] = item_h[row/12][k] * nb_h[row][k] ----
        for (int r = 0; r < 16; ++r) {
            const int R    = rowbase + r;
            const int item = R / S_NB;
            const float* ih = item_h + (size_t)item * DIM;
            const float* nb = nb_h + (size_t)R * DIM;
#pragma unroll
            for (int jj = 0; jj < 4; ++jj) {
                const int j = lane + jj * 32;
                apw[(j >> 1) * AP_STRIDE + r * 2 + (j & 1)] = ih[j] * nb[j];
            }
        }
        if (lane < 16) {                       // K=128 "feature": nb weight
            apw[64 * AP_STRIDE + lane * 2]     = nb_w[rowbase + lane];
            apw[64 * AP_STRIDE + lane * 2 + 1] = 0.0f;
        }
        __syncthreads();

        // ---- GEMM: [16 x 129] @ [129 x 128] via V_WMMA_F32_16X16X4_F32 ----
        v8f acc[8] = {};
        for (int k4 = 0; k4 <= 32; ++k4) {     // 33 K-steps (last = bias)
            const int k2 = k4 * 2 + half;
            v2f a = *(const v2f*)(apw + k2 * AP_STRIDE + 2 * l);
#pragma unroll
            for (int nt = 0; nt < 8; ++nt) {
                const int col = nt * 16 + l;
                v2f b = *(const v2f*)(w1t + k2 * W1T_STRIDE + 2 * col);
                acc[nt] = __builtin_amdgcn_wmma_f32_16x16x4_f32(
                    false, a, false, b, (short)0, acc[nt], false, false);
            }
        }

        // ---- logits: e[M] = sum_N lrelu(y[M,N]) * w2[N] ----
        float part[8] = {0.f, 0.f, 0.f, 0.f, 0.f, 0.f, 0.f, 0.f};
#pragma unroll
        for (int nt = 0; nt < 8; ++nt) {
            const float w2v = w2r[nt];
#pragma unroll
            for (int v = 0; v < 8; ++v) {
                float y = acc[nt][v];
                y = (y > 0.0f) ? y : 0.2f * y;  // leaky_relu 0.2
                part[v] += y * w2v;
            }
        }
        float e_my = 0.0f;
#pragma unroll
        for (int v = 0; v < 8; ++v) {           // reduce over 16-lane half
            float sv = part[v];
            sv += __shfl_xor(sv, 1, 32);
            sv += __shfl_xor(sv, 2, 32);
            sv += __shfl_xor(sv, 4, 32);
            sv += __shfl_xor(sv, 8, 32);
            if (l == v) e_my = sv;              // lane l holds M = l + 8*half
        }
        if (l < 8) evals[wave][t * 16 + half * 8 + l] = e_my;
        __syncthreads();
    }

    // ---- softmax over S=12 per item (lanes 0..3 each own one item) ----
    if (lane < IT_WAVE) {
        const float* ev = &evals[wave][lane * S_NB];
        float mx = ev[0];
#pragma unroll
        for (int ss = 1; ss < S_NB; ++ss) mx = fmaxf(mx, ev[ss]);
        float ex[S_NB];
        float sum = 0.0f;
#pragma unroll
        for (int ss = 0; ss < S_NB; ++ss) { ex[ss] = __expf(ev[ss] - mx); sum += ex[ss]; }
        const float inv = 1.0f / sum;
        float* al = &alph[wave][lane * S_NB];
#pragma unroll
        for (int ss = 0; ss < S_NB; ++ss) al[ss] = ex[ss] * inv;
    }
    __syncthreads();

    // ---- out[n] = sum_s alpha[n,s] * nb_h[n,s,:]  (coalesced) ----
#pragma unroll
    for (int i = 0; i < IT_WAVE; ++i) {
        const int item = item0 + i;
        const float* nb = nb_h + (size_t)item * S_NB * DIM;
        const float* al = &alph[wave][i * S_NB];
#pragma unroll
        for (int jj = 0; jj < 4; ++jj) {
            const int c = lane + jj * 32;
            float accv = 0.0f;
#pragma unroll
            for (int ss = 0; ss < S_NB; ++ss) accv += al[ss] * nb[(size_t)ss * DIM + c];
            outp[(size_t)item * DIM + c] = accv;
        }
    }
}

// ---------------------------------------------------------------------------
// final: agg = s0*agg0 + s1*agg1 ; w = sigmoid(hidden@w3 + agg@w4)
//        out = (1-w)*hidden + w*agg.   One block per row, 128 threads.
// ---------------------------------------------------------------------------
__global__ __launch_bounds__(128)
void final_mix_kernel(const float* __restrict__ hidden,
                      const float* __restrict__ agg0,
                      const float* __restrict__ agg1,
                      const float* __restrict__ w3,
                      const float* __restrict__ w4,
                      const float* __restrict__ s,
                      float* __restrict__ out)
{
    __shared__ float h_lds[DIM];
    __shared__ float a_lds[DIM];
    const int row = blockIdx.x;
    const int c   = threadIdx.x;
    const size_t base = (size_t)row * DIM;

    const float h = hidden[base + c];
    const float a = s[0] * agg0[base + c] + s[1] * agg1[base + c];
    h_lds[c] = h;
    a_lds[c] = a;
    __syncthreads();

    float z = 0.0f;
    for (int k = 0; k < DIM; ++k)
        z += h_lds[k] * w3[k * DIM + c] + a_lds[k] * w4[k * DIM + c];

    const float w = 1.0f / (1.0f + __expf(-z));
    out[base + c] = (1.0f - w) * h + w * a;
}

// ---------------------------------------------------------------------------
extern "C" void kernel_launch(void* const* d_in, const int* in_sizes, int n_in,
                              void* d_out, int out_size, void* d_ws, size_t ws_size,
                              hipStream_t stream) {
    (void)in_sizes; (void)n_in; (void)out_size; (void)ws_size;
    const float* hidden = (const float*)d_in[0];  // [64,50,128]
    const float* nh1    = (const float*)d_in[1];  // [64,600,128]
    const float* nh2    = (const float*)d_in[2];  // [64,7200,128]
    const float* nw0    = (const float*)d_in[3];  // [64,50,12]
    const float* nw1    = (const float*)d_in[4];  // [64,600,12]
    const float* w1     = (const float*)d_in[5];  // [129,128]
    const float* w2     = (const float*)d_in[6];  // [128,1]
    const float* w3     = (const float*)d_in[7];  // [128,128]
    const float* w4     = (const float*)d_in[8];  // [128,128]
    const float* s      = (const float*)d_in[9];  // [2,1]
    float* out = (float*)d_out;

    const int N0 = 64 * 50;     // 3200 items
    const int N1 = 64 * 600;    // 38400 items

    float* ws    = (float*)d_ws;
    float* agg01 = ws;                                  // [N1*128]
    float* agg00 = agg01 + (size_t)N1 * DIM;            // [N0*128]
    float* agg10 = agg00 + (size_t)N0 * DIM;            // [N0*128]
    float* w1t   = agg10 + (size_t)N0 * DIM;            // [66*256]

    prep_w1_kernel<<<66, 256, 0, stream>>>(w1, w1t);

    // hop 0
    agg_kernel<<<N0 / IT_WG, 128, 0, stream>>>(hidden, nh1, nw0, w1t, w2, agg00);
    agg_kernel<<<N1 / IT_WG, 128, 0, stream>>>(nh1, nh2, nw1, w1t, w2, agg01);
    // hop 1
    agg_kernel<<<N0 / IT_WG, 128, 0, stream>>>(agg00, agg01, nw0, w1t, w2, agg10);
    // gated mix
    final_mix_kernel<<<N0, 128, 0, stream>>>(hidden, agg00, agg10, w3, w4, s, out);
}